// HeteroCausalBeta_56581899157988
// MI455X (gfx1250) — compile-verified
//
#include <hip/hip_runtime.h>

// ---------------------------------------------------------------------------
// HeteroCausal 2-layer R-GCN for MI455X (gfx1250, wave32).
// GEMMs via V_WMMA_F32_16X16X4_F32 (fp32-exact; GEMM is <10% of runtime);
// edge scatter via pre-scaled global f32 atomics (L2-resident on 192MB L2).
// ---------------------------------------------------------------------------

typedef __attribute__((ext_vector_type(2))) float v2f;
typedef __attribute__((ext_vector_type(8))) float v8f;

#define NWN 50000
#define NTN 5000
#define NDN 50000
#define HIDN 128

// -------------------------- utility kernels --------------------------------

__global__ __launch_bounds__(256) void fill_zero_k(float* __restrict__ p, int n) {
  int i = blockIdx.x * 256 + threadIdx.x;
  if (i < n) p[i] = 0.0f;
}

__global__ __launch_bounds__(256) void relu_k(float* __restrict__ p, int n) {
  int i = blockIdx.x * 256 + threadIdx.x;
  if (i < n) p[i] = fmaxf(p[i], 0.0f);
}

__global__ __launch_bounds__(256) void count_k(const int* __restrict__ dst,
                                               float* __restrict__ cnt, int E) {
  int e = blockIdx.x * 256 + threadIdx.x;
  if (e < E) atomicAdd(&cnt[dst[e]], 1.0f);
}

__global__ __launch_bounds__(256) void inv_k(float* __restrict__ cnt, int n) {
  int i = blockIdx.x * 256 + threadIdx.x;
  if (i < n) cnt[i] = 1.0f / fmaxf(cnt[i], 1.0f);
}

// ftp = ft * (effect>0), ftn = ft * (effect<0); n = NT*128
__global__ __launch_bounds__(256) void mask_k(const float* __restrict__ ft,
                                              const float* __restrict__ eff,
                                              float* __restrict__ ftp,
                                              float* __restrict__ ftn, int n) {
  int i = blockIdx.x * 256 + threadIdx.x;
  if (i < n) {
    float e = eff[i >> 7];
    float v = ft[i];
    ftp[i] = (e > 0.0f) ? v : 0.0f;
    ftn[i] = (e < 0.0f) ? v : 0.0f;
  }
}

// -------------------------- WMMA fp32 GEMM ---------------------------------
// out[M,128] = alpha * A[M,K] @ W[K,128] (+ bias) (+ out if accflag)
// One wave -> 16 rows x full 128 cols (8 accumulators). 8 waves per block.
__global__ __launch_bounds__(256) void gemm_wmma_k(const float* __restrict__ A,
                                                   const float* __restrict__ W,
                                                   const float* __restrict__ bias,
                                                   float* __restrict__ out,
                                                   int M, int K, float alpha,
                                                   int accflag) {
  const int lane = threadIdx.x & 31;
  const int wave = threadIdx.x >> 5;
  const int half = lane >> 4;    // 0: lanes 0-15, 1: lanes 16-31
  const int l16  = lane & 15;
  const int m_base = (blockIdx.x * 8 + wave) * 16;
  if (m_base >= M) return;       // wave-uniform exit; EXEC stays all-1s below

  v8f acc[8];
#pragma unroll
  for (int i = 0; i < 8; ++i) acc[i] = (v8f){0,0,0,0,0,0,0,0};

  // Clamp the A row instead of predicating the load: an out-of-range A row
  // only feeds the same (out-of-range) D row, and stores are guarded below.
  const int row  = m_base + l16;
  const int rowc = (row < M) ? row : (M - 1);
  const float* Arow = A + (size_t)rowc * K;

  for (int k = 0; k < K; k += 4) {
    // A 16x4 f32 layout: VGPR0 = K (k+0 | k+2), VGPR1 = K (k+1 | k+3).
    // ka is even and rows are 512B-aligned -> single 8B global_load_b64.
    const int ka = k + half * 2;
    v2f a = *(const v2f*)(Arow + ka);
#pragma unroll
    for (int nt = 0; nt < 8; ++nt) {
      // B 4x16 layout: VGPR0 = rows (k+0 | k+2), VGPR1 = rows (k+1 | k+3)
      v2f b;
      b.x = W[ka * HIDN + nt * 16 + l16];
      b.y = W[(ka + 1) * HIDN + nt * 16 + l16];
      acc[nt] = __builtin_amdgcn_wmma_f32_16x16x4_f32(
          false, a, false, b, (short)0, acc[nt], false, false);
    }
  }

  // C/D layout: VGPR r holds rows m_base+r (lanes 0-15) and m_base+r+8
#pragma unroll
  for (int nt = 0; nt < 8; ++nt) {
    const int col = nt * 16 + l16;
    const float bv = bias ? bias[col] : 0.0f;
#pragma unroll
    for (int r = 0; r < 8; ++r) {
      const int rr = m_base + r + half * 8;
      if (rr < M) {
        float v = alpha * acc[nt][r] + bv;
        if (accflag) v += out[rr * HIDN + col];
        out[rr * HIDN + col] = v;
      }
    }
  }
}

// -------------------------- edge scatter -----------------------------------
// One wave per edge; lane handles 4 consecutive feature floats.
// acc[dst] += feat[src] * w[e] * inv_cnt[dst]   (mean via pre-scale).
// Edge metadata is wave-uniform: force it scalar so src/dst/w/inv become
// SMEM loads and the vector pipe is left to the f32 atomics.
__global__ __launch_bounds__(256) void scatter_k(const int* __restrict__ src,
                                                 const int* __restrict__ dst,
                                                 const float* __restrict__ w,
                                                 const float* __restrict__ inv,
                                                 const float* __restrict__ feat,
                                                 float* __restrict__ acc, int E) {
  int e = blockIdx.x * 8 + (threadIdx.x >> 5);
  const int lane = threadIdx.x & 31;
  if (e >= E) return;
  e = __builtin_amdgcn_readfirstlane(e);
  const int s = __builtin_amdgcn_readfirstlane(src[e]);
  const int d = __builtin_amdgcn_readfirstlane(dst[e]);
  const float scale = w[e] * inv[d];
  const float4* fs = (const float4*)(feat + (size_t)s * HIDN);
  float4 v = fs[lane];
  float* ap = acc + (size_t)d * HIDN + lane * 4;
  atomicAdd(ap + 0, v.x * scale);
  atomicAdd(ap + 1, v.y * scale);
  atomicAdd(ap + 2, v.z * scale);
  atomicAdd(ap + 3, v.w * scale);
}

// ---------------------------------------------------------------------------

extern "C" void kernel_launch(void* const* d_in, const int* in_sizes, int n_in,
                              void* d_out, int out_size, void* d_ws, size_t ws_size,
                              hipStream_t stream) {
  const float* feat_word  = (const float*)d_in[0];
  const float* feat_topic = (const float*)d_in[1];
  const float* effect     = (const float*)d_in[2];
  const int   *ww_src = (const int*)d_in[3],  *ww_dst = (const int*)d_in[4];
  const float *ww_w   = (const float*)d_in[5];
  const int   *wt_src = (const int*)d_in[6],  *wt_dst = (const int*)d_in[7];
  const float *wt_w   = (const float*)d_in[8];
  const int   *wd_src = (const int*)d_in[9],  *wd_dst = (const int*)d_in[10];
  const float *wd_w   = (const float*)d_in[11];
  const int   *tt_src = (const int*)d_in[12], *tt_dst = (const int*)d_in[13];
  const float *tt_w   = (const float*)d_in[14];
  const int   *td_src = (const int*)d_in[15], *td_dst = (const int*)d_in[16];
  const float *td_w   = (const float*)d_in[17];
  const int E_ww = in_sizes[3], E_wt = in_sizes[6], E_wd = in_sizes[9];
  const int E_tt = in_sizes[12], E_td = in_sizes[15];

  const float *l1_ww_W = (const float*)d_in[18], *l1_ww_b = (const float*)d_in[19];
  const float *l1_wt_W = (const float*)d_in[20], *l1_wt_b = (const float*)d_in[21];
  const float *l1_tt_W = (const float*)d_in[24], *l1_tt_b = (const float*)d_in[25];
  const float *l1_cau_W = (const float*)d_in[28], *l1_noi_W = (const float*)d_in[29];
  const float *l2_ww_W = (const float*)d_in[30], *l2_ww_b = (const float*)d_in[31];
  const float *l2_wt_W = (const float*)d_in[32], *l2_wt_b = (const float*)d_in[33];
  const float *l2_wd_W = (const float*)d_in[34], *l2_wd_b = (const float*)d_in[35];
  const float *l2_tt_W = (const float*)d_in[36], *l2_tt_b = (const float*)d_in[37];
  const float *l2_td_W = (const float*)d_in[38], *l2_td_b = (const float*)d_in[39];
  const float *l2_cau_W = (const float*)d_in[40], *l2_noi_W = (const float*)d_in[41];

  // ---- workspace carve (all 512B-aligned slabs first) ----
  float* ws = (float*)d_ws;
  size_t off = 0;
  float* A_ww = ws + off; off += (size_t)NWN * HIDN;
  float* A_wt = ws + off; off += (size_t)NWN * HIDN;
  float* A_wd = ws + off; off += (size_t)NWN * HIDN;
  float* A_tt = ws + off; off += (size_t)NTN * HIDN;
  float* A_td = ws + off; off += (size_t)NTN * HIDN;
  float* hw   = ws + off; off += (size_t)NWN * HIDN;
  float* ht   = ws + off; off += (size_t)NTN * HIDN;
  float* ftp  = ws + off; off += (size_t)NTN * HIDN;
  float* ftn  = ws + off; off += (size_t)NTN * HIDN;
  float* inv_ww = ws + off; off += NWN;
  float* inv_wt = ws + off; off += NTN;
  float* inv_tt = ws + off; off += NTN;
  float* inv_wd = ws + off; off += NDN;
  float* inv_td = ws + off; off += NDN;
  if (ws_size < off * sizeof(float)) return;

  auto cdiv = [](int a, int b) { return (a + b - 1) / b; };

  auto gemm = [&](const float* A, const float* W, const float* b, float* o,
                  int M, int K, float alpha, int acc) {
    gemm_wmma_k<<<cdiv(M, 128), 256, 0, stream>>>(A, W, b, o, M, K, alpha, acc);
  };
  auto scatter = [&](const int* s, const int* d, const float* w, const float* inv,
                     const float* feat, float* acc, int E) {
    scatter_k<<<cdiv(E, 8), 256, 0, stream>>>(s, d, w, inv, feat, acc, E);
  };
  auto prep_inv = [&](const int* d, float* inv, int n, int E) {
    fill_zero_k<<<cdiv(n, 256), 256, 0, stream>>>(inv, n);
    count_k<<<cdiv(E, 256), 256, 0, stream>>>(d, inv, E);
    inv_k<<<cdiv(n, 256), 256, 0, stream>>>(inv, n);
  };

  // ---- per-relation 1/max(indeg,1), shared by both layers ----
  prep_inv(ww_dst, inv_ww, NWN, E_ww);
  prep_inv(wt_dst, inv_wt, NTN, E_wt);
  prep_inv(tt_dst, inv_tt, NTN, E_tt);
  prep_inv(wd_dst, inv_wd, NDN, E_wd);
  prep_inv(td_dst, inv_td, NDN, E_td);

  const int NTH = NTN * HIDN;

  // =================== layer 1 (hd is dead -> skip wd/td) ===================
  mask_k<<<cdiv(NTH, 256), 256, 0, stream>>>(feat_topic, effect, ftp, ftn, NTH);

  gemm(feat_word, l1_ww_W, l1_ww_b, A_ww, NWN, 256, 1.0f, 0);
  gemm(feat_word, l1_wt_W, l1_wt_b, A_wt, NWN, 256, 1.0f, 0);
  gemm(feat_topic, l1_tt_W, l1_tt_b, A_tt, NTN, 128, 1.0f, 0);
  gemm(ftp, l1_cau_W, nullptr, A_tt, NTN, 128, 1.0f, 1);   // + (ft*pos)@cau
  gemm(ftn, l1_noi_W, nullptr, A_tt, NTN, 128, -1.0f, 1);  // - (ft*neg)@noi

  fill_zero_k<<<cdiv(NWN * HIDN, 256), 256, 0, stream>>>(hw, NWN * HIDN);
  fill_zero_k<<<cdiv(NTH, 256), 256, 0, stream>>>(ht, NTH);
  scatter(ww_src, ww_dst, ww_w, inv_ww, A_ww, hw, E_ww);
  scatter(wt_src, wt_dst, wt_w, inv_wt, A_wt, ht, E_wt);
  scatter(tt_src, tt_dst, tt_w, inv_tt, A_tt, ht, E_tt);
  relu_k<<<cdiv(NWN * HIDN, 256), 256, 0, stream>>>(hw, NWN * HIDN);
  relu_k<<<cdiv(NTH, 256), 256, 0, stream>>>(ht, NTH);

  // =================== layer 2 ===================
  mask_k<<<cdiv(NTH, 256), 256, 0, stream>>>(ht, effect, ftp, ftn, NTH);

  gemm(hw, l2_ww_W, l2_ww_b, A_ww, NWN, 128, 1.0f, 0);
  gemm(hw, l2_wt_W, l2_wt_b, A_wt, NWN, 128, 1.0f, 0);
  gemm(hw, l2_wd_W, l2_wd_b, A_wd, NWN, 128, 1.0f, 0);
  gemm(ht, l2_tt_W, l2_tt_b, A_tt, NTN, 128, 1.0f, 0);
  gemm(ftp, l2_cau_W, nullptr, A_tt, NTN, 128, 1.0f, 1);
  gemm(ftn, l2_noi_W, nullptr, A_tt, NTN, 128, -1.0f, 1);
  gemm(ht, l2_td_W, l2_td_b, A_td, NTN, 128, 1.0f, 0);
  gemm(ftp, l2_cau_W, nullptr, A_td, NTN, 128, 1.0f, 1);
  gemm(ftn, l2_noi_W, nullptr, A_td, NTN, 128, -1.0f, 1);

  float* out   = (float*)d_out;
  float* out_w = out;                                   // hw2 [NW,128]
  float* out_t = out + (size_t)NWN * HIDN;              // ht2 [NT,128]
  float* out_d = out_t + (size_t)NTN * HIDN;            // hd2 [ND,128]
  fill_zero_k<<<cdiv(out_size, 256), 256, 0, stream>>>(out, out_size);
  scatter(ww_src, ww_dst, ww_w, inv_ww, A_ww, out_w, E_ww);
  scatter(wt_src, wt_dst, wt_w, inv_wt, A_wt, out_t, E_wt);
  scatter(tt_src, tt_dst, tt_w, inv_tt, A_tt, out_t, E_tt);
  scatter(wd_src, wd_dst, wd_w, inv_wd, A_wd, out_d, E_wd);
  scatter(td_src, td_dst, td_w, inv_td, A_td, out_d, E_td);
  relu_k<<<cdiv(out_size, 256), 256, 0, stream>>>(out, out_size);
}